// FBPINN_30940944401057
// MI455X (gfx1250) — compile-verified
//
#include <hip/hip_runtime.h>
#include <math.h>

typedef __attribute__((ext_vector_type(2))) float v2f;
typedef __attribute__((ext_vector_type(8))) float v8f;

#if __has_builtin(__builtin_amdgcn_tanhf)
__device__ __forceinline__ float fast_tanh(float x) { return __builtin_amdgcn_tanhf(x); }
#elif __has_builtin(__builtin_amdgcn_tanh_f32)
__device__ __forceinline__ float fast_tanh(float x) { return __builtin_amdgcn_tanh_f32(x); }
#else
__device__ __forceinline__ float fast_tanh(float x) { return tanhf(x); }
#endif

__device__ __forceinline__ float sat01(float v) { return fminf(fmaxf(v, 0.0f), 1.0f); }

#define NPTS   32768
#define S_SUB  32
#define WID    64
#define PAD    68          // LDS row stride in floats (68 mod 64 == 4 -> no bank conflicts)
#define WAVES  4
#define TPB    (WAVES * 32)
#define PTS_PER_BLOCK (WAVES * 16)
#define TRANS_INV 10.0f    // 1/0.1
#define PI_F 3.14159265358979f

__global__ __launch_bounds__(TPB) void fbpinn_kernel(
    const float* __restrict__ gx,
    const float* __restrict__ gxmins, const float* __restrict__ gxmaxs,
    const float* __restrict__ gW0, const float* __restrict__ gb0,
    const float* __restrict__ gW1, const float* __restrict__ gb1,
    const float* __restrict__ gW2, const float* __restrict__ gb2,
    const float* __restrict__ gWo, const float* __restrict__ gbo,
    float* __restrict__ gout)
{
    __shared__ float sW1[WID * PAD];              // [k_out][h_in], padded rows
    __shared__ float sW2[WID * PAD];
    __shared__ float sAct[WAVES][16 * PAD];       // per-wave 16x64 activations, padded
    __shared__ float sW0[WID * 2];
    __shared__ float sB0[WID], sB1[WID], sB2[WID], sWo[WID];

    const int tid  = threadIdx.x;
    const int wv   = tid >> 5;
    const int lane = tid & 31;
    const int m    = lane & 15;     // point row within the wave's 16-point tile
    const int half = lane >> 4;     // 0 or 1 (which 16-lane half)

    const int pt = blockIdx.x * PTS_PER_BLOCK + wv * 16 + m;
    const float x0 = gx[pt * 2 + 0];
    const float x1 = gx[pt * 2 + 1];

    float num = 0.0f, wsum = 0.0f;
    float* act = sAct[wv];

    for (int s = 0; s < S_SUB; ++s) {
        __syncthreads();   // protect previous iteration's readers before overwrite
        // ---- stage weights for subdomain s into LDS ----
        for (int i = tid; i < WID * WID; i += TPB) {
            const int r = i >> 6, c = i & 63;
            sW1[r * PAD + c] = gW1[s * WID * WID + i];
            sW2[r * PAD + c] = gW2[s * WID * WID + i];
        }
        if (tid < WID) {
            sB0[tid] = gb0[s * WID + tid];
            sB1[tid] = gb1[s * WID + tid];
            sB2[tid] = gb2[s * WID + tid];
            sWo[tid] = gWo[s * WID + tid];
            sW0[tid * 2 + 0] = gW0[s * WID * 2 + tid * 2 + 0];
            sW0[tid * 2 + 1] = gW0[s * WID * 2 + tid * 2 + 1];
        }
        __syncthreads();

        // ---- subdomain box / normalized input ----
        const float mn0 = gxmins[s * 2 + 0], mn1 = gxmins[s * 2 + 1];
        const float mx0 = gxmaxs[s * 2 + 0], mx1 = gxmaxs[s * 2 + 1];
        const float c0 = 0.5f * (mn0 + mx0), c1 = 0.5f * (mn1 + mx1);
        const float sc0 = fmaxf(0.5f * (mx0 - mn0), 1e-9f);
        const float sc1 = fmaxf(0.5f * (mx1 - mn1), 1e-9f);
        const float xn0 = (x0 - c0) / sc0;
        const float xn1 = (x1 - c1) / sc1;

        // ---- layer 0: 2 -> 64 (scalar; each lane does 32 hidden units of point m) ----
        {
            const int jbase = half * 32;
            #pragma unroll
            for (int i = 0; i < 32; ++i) {
                const int j = jbase + i;
                const float h = fast_tanh(xn0 * sW0[j * 2 + 0] + xn1 * sW0[j * 2 + 1] + sB0[j]);
                act[m * PAD + j] = h;
            }
        }
        // same-wave LDS ops are in-order; each wave only touches its own act slice.

        // ---- layers 1 & 2: 64 -> 64 via V_WMMA_F32_16X16X4_F32 ----
        #pragma unroll
        for (int layer = 0; layer < 2; ++layer) {
            const float* Wl = (layer == 0) ? sW1 : sW2;
            const float* Bl = (layer == 0) ? sB1 : sB2;

            v8f acc0 = {}, acc1 = {}, acc2 = {}, acc3 = {};
            #pragma unroll
            for (int kb = 0; kb < 16; ++kb) {
                const int krow = kb * 4 + 2 * half;  // K index pair {krow, krow+1}
                // A fragment: A[m][krow], A[m][krow+1]
                const v2f a = *(const v2f*)&act[m * PAD + krow];
                // B fragments: B[k][n] = Wl[n][k], n = nb*16 + m
                const v2f b0 = *(const v2f*)&Wl[(0 * 16 + m) * PAD + krow];
                const v2f b1 = *(const v2f*)&Wl[(1 * 16 + m) * PAD + krow];
                const v2f b2 = *(const v2f*)&Wl[(2 * 16 + m) * PAD + krow];
                const v2f b3 = *(const v2f*)&Wl[(3 * 16 + m) * PAD + krow];
                acc0 = __builtin_amdgcn_wmma_f32_16x16x4_f32(false, a, false, b0, (short)0, acc0, false, false);
                acc1 = __builtin_amdgcn_wmma_f32_16x16x4_f32(false, a, false, b1, (short)0, acc1, false, false);
                acc2 = __builtin_amdgcn_wmma_f32_16x16x4_f32(false, a, false, b2, (short)0, acc2, false, false);
                acc3 = __builtin_amdgcn_wmma_f32_16x16x4_f32(false, a, false, b3, (short)0, acc3, false, false);
            }
            // bias + tanh + store back (all A-reads above precede these writes; wave LDS is in-order)
            v8f accs[4] = {acc0, acc1, acc2, acc3};
            #pragma unroll
            for (int nb = 0; nb < 4; ++nb) {
                const int col = nb * 16 + m;       // N index of this lane
                const float bias = Bl[col];
                #pragma unroll
                for (int j = 0; j < 8; ++j) {
                    const int row = j + 8 * half;  // M index of VGPR j in this lane half
                    act[row * PAD + col] = fast_tanh(accs[nb][j] + bias);
                }
            }
        }

        // ---- output layer (64 -> 1) + window, lanes 0..15 only ----
        if (half == 0) {
            float dot = 0.0f;
            #pragma unroll
            for (int j = 0; j < WID; ++j) dot += act[m * PAD + j] * sWo[j];
            dot += gbo[s];

            const float tu0 = sat01((x0 - mn0) * TRANS_INV);
            const float td0 = sat01((mx0 - x0) * TRANS_INV);
            const float tu1 = sat01((x1 - mn1) * TRANS_INV);
            const float td1 = sat01((mx1 - x1) * TRANS_INV);
            const float pd0 = 0.25f * (1.0f - __cosf(PI_F * tu0)) * (1.0f - __cosf(PI_F * td0));
            const float pd1 = 0.25f * (1.0f - __cosf(PI_F * tu1)) * (1.0f - __cosf(PI_F * td1));
            const float w = pd0 * pd1;

            num  += dot * w;
            wsum += w;
        }
    }

    if (half == 0) {
        gout[pt] = num / (wsum + 1e-8f);
    }
}

extern "C" void kernel_launch(void* const* d_in, const int* in_sizes, int n_in,
                              void* d_out, int out_size, void* d_ws, size_t ws_size,
                              hipStream_t stream) {
    (void)in_sizes; (void)n_in; (void)out_size; (void)d_ws; (void)ws_size;
    const float* gx     = (const float*)d_in[0];
    const float* gxmins = (const float*)d_in[1];
    const float* gxmaxs = (const float*)d_in[2];
    const float* gW0    = (const float*)d_in[3];
    const float* gb0    = (const float*)d_in[4];
    const float* gW1    = (const float*)d_in[5];
    const float* gb1    = (const float*)d_in[6];
    const float* gW2    = (const float*)d_in[7];
    const float* gb2    = (const float*)d_in[8];
    const float* gWo    = (const float*)d_in[9];
    const float* gbo    = (const float*)d_in[10];

    const int blocks = NPTS / PTS_PER_BLOCK;  // 32768 / 64 = 512
    fbpinn_kernel<<<blocks, TPB, 0, stream>>>(gx, gxmins, gxmaxs,
                                              gW0, gb0, gW1, gb1, gW2, gb2, gWo, gbo,
                                              (float*)d_out);
}